// GeneralizedMultiCSLoss_50225347560091
// MI455X (gfx1250) — compile-verified
//
#include <hip/hip_runtime.h>
#include <math.h>

typedef __attribute__((ext_vector_type(2))) float v2f;
typedef __attribute__((ext_vector_type(8))) float v8f;

#define NPTS 8192
#define DK   64      // inner dimension
#define TS   64      // output tile side per block
#define LDP  68      // padded LDS row stride (floats) -> conflict-free b64 reads

// job -> (A index, Z index):  0:(0,0) 1:(1,1) 2:(2,2) 3:(0,1) 4:(0,2) 5:(1,2)
__constant__ int c_ai[6] = {0, 1, 2, 0, 0, 1};
__constant__ int c_bi[6] = {0, 1, 2, 1, 2, 2};

__global__ void mcs_zero_kernel(double* sums) {
    if (threadIdx.x < 6) sums[threadIdx.x] = 0.0;
}

__global__ __launch_bounds__(256)
void mcs_gram_kernel(const float* __restrict__ p0,
                     const float* __restrict__ p1,
                     const float* __restrict__ p2,
                     double* __restrict__ sums)
{
    __shared__ float Xs[TS][LDP];
    __shared__ float Zs[TS][LDP];
    __shared__ float x2s[TS];
    __shared__ float z2s[TS];
    __shared__ float red[256];

    const int job = blockIdx.z;
    const int bx  = blockIdx.x;
    const int by  = blockIdx.y;

    // Self Grams are symmetric: skip strict lower triangle, weight upper x2.
    if (job < 3 && by > bx) return;

    const float* bases[3] = {p0, p1, p2};
    const float* __restrict__ X = bases[c_ai[job]] + (size_t)by * TS * DK;
    const float* __restrict__ Z = bases[c_bi[job]] + (size_t)bx * TS * DK;

    const int t = threadIdx.x;

    // ---- Stage 64x64 X and Z tiles into LDS (coalesced float4 loads) ----
    #pragma unroll
    for (int i = 0; i < 4; ++i) {
        int lin = t + 256 * i;          // 0..1023 float4 slots per tile
        int row = lin >> 4;             // 16 float4 per row
        int c4  = (lin & 15) << 2;      // float column
        float4 vx = *(const float4*)(X + row * DK + c4);
        Xs[row][c4 + 0] = vx.x; Xs[row][c4 + 1] = vx.y;
        Xs[row][c4 + 2] = vx.z; Xs[row][c4 + 3] = vx.w;
        float4 vz = *(const float4*)(Z + row * DK + c4);
        Zs[row][c4 + 0] = vz.x; Zs[row][c4 + 1] = vz.y;
        Zs[row][c4 + 2] = vz.z; Zs[row][c4 + 3] = vz.w;
    }
    __syncthreads();

    // ---- Row squared norms ----
    if (t < TS) {
        float s = 0.f;
        #pragma unroll 8
        for (int k = 0; k < DK; ++k) { float v = Xs[t][k]; s += v * v; }
        x2s[t] = s;
    } else if (t < 2 * TS) {
        int r = t - TS;
        float s = 0.f;
        #pragma unroll 8
        for (int k = 0; k < DK; ++k) { float v = Zs[r][k]; s += v * v; }
        z2s[r] = s;
    }
    __syncthreads();

    // ---- 8 waves in a 4x2 arrangement; each wave: two 16x16 WMMA subtiles ----
    const int wave = t >> 5;
    const int lane = t & 31;
    const int wr = wave & 3;            // subtile row    (0..3)
    const int wc = wave >> 2;           // subtile column (0..1), covers 2x16 cols
    const int lh = lane >> 4;           // lane half: K offset {0,2}
    const int ml = lane & 15;           // matrix row/col within subtile

    const int mbase = wr * 16;
    const float* __restrict__ xrow = &Xs[mbase + ml][0];

    float part = 0.f;

    #pragma unroll
    for (int s = 0; s < 2; ++s) {
        const int nbase = wc * 32 + s * 16;
        const float* __restrict__ zrow = &Zs[nbase + ml][0];

        v8f acc = {0.f, 0.f, 0.f, 0.f, 0.f, 0.f, 0.f, 0.f};
        #pragma unroll
        for (int k0 = 0; k0 < DK; k0 += 4) {
            // A 16x4 fp32 fragment: lane half selects K pair {0,1}/{2,3}
            v2f a = *(const v2f*)(xrow + k0 + 2 * lh);
            // B 4x16 fp32 fragment: same per-lane pattern (Z row-major = B^T)
            v2f b = *(const v2f*)(zrow + k0 + 2 * lh);
            acc = __builtin_amdgcn_wmma_f32_16x16x4_f32(
                false, a, false, b, (short)0, acc, false, false);
        }

        // C layout: n = nbase + ml ; m = mbase + r + 8*lh for VGPR r
        const float z2 = z2s[nbase + ml];
        #pragma unroll
        for (int r = 0; r < 8; ++r) {
            float x2  = x2s[mbase + r + 8 * lh];
            float g   = x2 + z2 - 2.0f * acc[r];
            part += __expf(g * (-1.0f / 128.0f));   // exp(-G/KSIZE)
        }
    }

    // ---- Deterministic in-block reduction, then one f64 atomic per block ----
    red[t] = part;
    __syncthreads();
    #pragma unroll
    for (int off = 128; off > 0; off >>= 1) {
        if (t < off) red[t] += red[t + off];
        __syncthreads();
    }
    if (t == 0) {
        double w = (job < 3 && by < bx) ? 2.0 : 1.0;
        atomicAdd(&sums[job], (double)red[0] * w);
    }
}

__global__ void mcs_finalize_kernel(const double* __restrict__ sums,
                                    float* __restrict__ out)
{
    const double NN   = (double)NPTS * (double)NPTS;
    const double norm = sqrt(2.0 * 3.14159265358979323846 * 128.0);
    double m[6];
    #pragma unroll
    for (int i = 0; i < 6; ++i) m[i] = sums[i] / (NN * norm);

    const int ii[3] = {0, 0, 1};
    const int jj[3] = {1, 2, 2};
    const int cc[3] = {3, 4, 5};
    double loss = 0.0;
    #pragma unroll
    for (int p = 0; p < 3; ++p) {
        loss += log(3.0 * sqrt(m[ii[p]] * m[jj[p]] + 1e-5) / (m[cc[p]] + 1e-5));
    }
    out[0] = (float)loss;
}

extern "C" void kernel_launch(void* const* d_in, const int* in_sizes, int n_in,
                              void* d_out, int out_size, void* d_ws, size_t ws_size,
                              hipStream_t stream) {
    (void)in_sizes; (void)n_in; (void)out_size; (void)ws_size;
    const float* p0 = (const float*)d_in[0];
    const float* p1 = (const float*)d_in[1];
    const float* p2 = (const float*)d_in[2];
    double* sums = (double*)d_ws;   // 6 doubles of scratch

    mcs_zero_kernel<<<1, 32, 0, stream>>>(sums);

    dim3 grid(NPTS / TS, NPTS / TS, 6);   // 128 x 128 x 6
    mcs_gram_kernel<<<grid, 256, 0, stream>>>(p0, p1, p2, sums);

    mcs_finalize_kernel<<<1, 1, 0, stream>>>(sums, (float*)d_out);
}